// Quantizer_69621419868276
// MI455X (gfx1250) — compile-verified
//
#include <hip/hip_runtime.h>
#include <hip/hip_bf16.h>

typedef float v2f __attribute__((ext_vector_type(2)));
typedef float v8f __attribute__((ext_vector_type(8)));

#define K_CODES 512
#define LATENT 12
#define BATCH_ 512
#define TOKENS_ 256
#define TOTAL_TOKENS (BATCH_ * TOKENS_)   // 131072
#define TOKENS_PER_WAVE 16
#define WAVES_PER_BLOCK 8
#define THREADS_PER_BLOCK (WAVES_PER_BLOCK * 32)
#define TOKENS_PER_BLOCK (WAVES_PER_BLOCK * TOKENS_PER_WAVE) // 128
#define CB_STRIDE 18   // padded row: [-2w(12) | w2 | 0 0 0 | pad pad]; 72B -> conflict-free

// One wave handles 16 tokens x all 512 codes. Distances come straight out of
// V_WMMA_F32_16X16X4_F32 by augmenting K: A' = [x, 1, 0,0,0], B' = [-2w, w2, 0,0,0],
// so acc = -2*x.w + w2 (the ||x||^2 term is constant over the argmin axis).
__global__ __launch_bounds__(THREADS_PER_BLOCK)
void vq_argmin_wmma_kernel(const float* __restrict__ x,
                           const float* __restrict__ cb,
                           float* __restrict__ out_q,
                           float* __restrict__ out_idx,
                           float* __restrict__ out_loss)
{
    __shared__ float lds_cbx[K_CODES * CB_STRIDE];   // 36 KB

    const int tid = threadIdx.x;

    // Stage augmented codebook: row k = [-2*w0..-2*w11, ||w||^2, 0,0,0, pad,pad]
    for (int k = tid; k < K_CODES; k += THREADS_PER_BLOCK) {
        float w2 = 0.f;
        float* dst = &lds_cbx[k * CB_STRIDE];
        #pragma unroll
        for (int d = 0; d < LATENT; ++d) {
            const float w = cb[k * LATENT + d];
            w2 += w * w;
            dst[d] = -2.0f * w;
        }
        dst[LATENT] = w2;
        #pragma unroll
        for (int d = LATENT + 1; d < CB_STRIDE; ++d) dst[d] = 0.f;
    }
    __syncthreads();

    const int lane  = tid & 31;
    const int wave  = tid >> 5;
    const int token_base = (blockIdx.x * WAVES_PER_BLOCK + wave) * TOKENS_PER_WAVE;

    const int half  = lane >> 4;      // 0: lanes 0-15, 1: lanes 16-31
    const int l15   = lane & 15;
    const int khalf = half * 2;       // 32-bit A/B layout: lanes 16-31 hold K+2,K+3

    // A-matrix (16 tokens x 16 dims) in WMMA f32 layout, 4 chunks of K=4.
    // Chunks 0-2: raw x; chunk 3 covers dims 12..15 = [1, 0, 0, 0].
    v2f A[4];
    const float* xrow = x + (size_t)(token_base + l15) * LATENT;
    #pragma unroll
    for (int c = 0; c < 3; ++c) {
        A[c].x = xrow[4 * c + khalf + 0];
        A[c].y = xrow[4 * c + khalf + 1];
    }
    A[3].x = (half == 0) ? 1.0f : 0.0f;   // dim 12 (half0) / dim 14 (half1)
    A[3].y = 0.0f;                        // dim 13 / dim 15

    // Running argmin; C/D layout: lane,vgpr r -> (M = r + 8*half, N = l15).
    float best_val[8];
    int   best_idx[8];
    #pragma unroll
    for (int r = 0; r < 8; ++r) { best_val[r] = 3.4e38f; best_idx[r] = 0; }

    #pragma unroll 4
    for (int t = 0; t < K_CODES / 16; ++t) {
        const int code = t * 16 + l15;
        const float* crow = &lds_cbx[code * CB_STRIDE + khalf];
        v8f acc = {};
        #pragma unroll
        for (int c = 0; c < 4; ++c) {
            v2f B;
            B.x = crow[4 * c + 0];
            B.y = crow[4 * c + 1];
            acc = __builtin_amdgcn_wmma_f32_16x16x4_f32(
                /*neg_a=*/false, A[c], /*neg_b=*/false, B,
                /*c_mod=*/(short)0, acc, /*reuse_a=*/false, /*reuse_b=*/false);
        }
        #pragma unroll
        for (int r = 0; r < 8; ++r) {
            const float dist = acc[r];            // already w2 - 2*x.w
            if (dist < best_val[r]) { best_val[r] = dist; best_idx[r] = code; }
        }
    }

    // Cross-lane argmin within each 16-lane half (rows are disjoint per half).
    // Tie-break on smaller index to match jnp.argmin.
    #pragma unroll
    for (int r = 0; r < 8; ++r) {
        float v = best_val[r]; int ix = best_idx[r];
        #pragma unroll
        for (int m = 8; m >= 1; m >>= 1) {
            float ov = __shfl_xor(v, m, 32);
            int   oi = __shfl_xor(ix, m, 32);
            if (ov < v || (ov == v && oi < ix)) { v = ov; ix = oi; }
        }
        best_val[r] = v; best_idx[r] = ix;
    }

    // Emit quantized vectors (recover w = -0.5 * (-2w)), indices, loss partial.
    float lsum = 0.f;
    #pragma unroll
    for (int r = 0; r < 8; ++r) {
        const int row   = r + 8 * half;         // 0..15
        const int token = token_base + row;
        const int ix    = best_idx[r];          // uniform across this 16-lane half
        if (l15 < LATENT) {
            const float q  = -0.5f * lds_cbx[ix * CB_STRIDE + l15];
            out_q[(size_t)token * LATENT + l15] = q;
            const float xv = x[(size_t)token * LATENT + l15];
            const float dd = q - xv;
            lsum += dd * dd;
        }
        if (l15 == 0) out_idx[token] = (float)ix;
    }
    // Wave-wide sum, then one atomic per wave.
    #pragma unroll
    for (int m = 16; m >= 1; m >>= 1) lsum += __shfl_xor(lsum, m, 32);
    if (lane == 0) {
        const float scale = 1.25f / (float)((size_t)TOTAL_TOKENS * LATENT);
        atomicAdd(out_loss, lsum * scale);
    }
}

extern "C" void kernel_launch(void* const* d_in, const int* in_sizes, int n_in,
                              void* d_out, int out_size, void* d_ws, size_t ws_size,
                              hipStream_t stream) {
    const float* x  = (const float*)d_in[0];   // [512, 256, 12]
    const float* cb = (const float*)d_in[1];   // [512, 12]

    float* out      = (float*)d_out;
    float* out_q    = out;                                     // [B,N,D]
    float* out_idx  = out + (size_t)TOTAL_TOKENS * LATENT;     // [B,N] as float
    float* out_loss = out_idx + TOTAL_TOKENS;                  // scalar

    // Zero the loss accumulator (graph-capture-safe async memset).
    hipMemsetAsync(out_loss, 0, sizeof(float), stream);

    const int blocks = TOTAL_TOKENS / TOKENS_PER_BLOCK;        // 1024
    vq_argmin_wmma_kernel<<<blocks, THREADS_PER_BLOCK, 0, stream>>>(
        x, cb, out_q, out_idx, out_loss);
}